// Gate_63350767616767
// MI455X (gfx1250) — compile-verified
//
#include <hip/hip_runtime.h>
#include <hip/hip_bf16.h>
#include <math.h>

// ---------------------------------------------------------------------------
// MoE gate (DeepSeek-V3 style) for MI455X / gfx1250.
//
// Kernel 1: scores = sigmoid(x @ W^T), exact fp32 via V_WMMA_F32_16X16X4_F32.
//           Register-blocked: one wave32 computes a 32x32 (tokens x experts)
//           tile = 4 independent 16x16 accumulators. Per K-step (K=4):
//           2 A-loads + 2 B-loads feed 4 WMMAs (1.0 load/WMMA, ILP=4).
// Kernel 2: grouped top-k routing, one wave32 per token.
// ---------------------------------------------------------------------------

#define DIM        4096
#define N_EXPERTS  256
#define TOPK       8
#define N_GROUPS   8
#define TOPK_GROUPS 4
#define ROUTE_SCALE 2.5f

typedef __attribute__((ext_vector_type(2))) float v2f;
typedef __attribute__((ext_vector_type(8))) float v8f;

static __device__ __forceinline__ v8f wmma_f32(v2f a, v2f b, v8f c) {
    return __builtin_amdgcn_wmma_f32_16x16x4_f32(
        false, a, false, b, (short)0, c, false, false);
}

// ---------------------------------------------------------------------------
// GEMM + sigmoid. grid = nTokens/32, block = 256 (8 waves).
// Wave w handles experts [32w, 32w+32); all waves share the 32-token tile.
// ---------------------------------------------------------------------------
__global__ __launch_bounds__(256)
void gate_gemm_sigmoid(const float* __restrict__ x,
                       const float* __restrict__ w,
                       float* __restrict__ scores)
{
    const int lane = threadIdx.x & 31;
    const int wave = threadIdx.x >> 5;

    const int m0 = blockIdx.x * 32;        // token block base (32 tokens)
    const int n0 = wave * 32;              // expert block base (32 experts)

    const int half = lane >> 4;            // 0: K{0,1}, 1: K{2,3}
    const int l    = lane & 15;

    // A: 16x4 f32 -> lane holds float2 at x[row][k + 2*half]
    const float* a0 = x + (size_t)(m0 + l) * DIM + 2 * half;
    const float* a1 = a0 + (size_t)16 * DIM;
    // B: 4x16 f32, B = W^T -> lane holds float2 at w[col][k + 2*half]
    const float* b0 = w + (size_t)(n0 + l) * DIM + 2 * half;
    const float* b1 = b0 + (size_t)16 * DIM;

    v8f c00 = {}, c01 = {}, c10 = {}, c11 = {};

    for (int k = 0; k < DIM; k += 16) {
#pragma unroll
        for (int kk = 0; kk < 16; kk += 4) {
            v2f av0 = *(const v2f*)(a0 + k + kk);
            v2f av1 = *(const v2f*)(a1 + k + kk);
            v2f bv0 = *(const v2f*)(b0 + k + kk);
            v2f bv1 = *(const v2f*)(b1 + k + kk);
            c00 = wmma_f32(av0, bv0, c00);   // 4 independent acc chains
            c01 = wmma_f32(av0, bv1, c01);
            c10 = wmma_f32(av1, bv0, c10);
            c11 = wmma_f32(av1, bv1, c11);
        }
    }

    // sigmoid + store. D layout: VGPR r -> row +r + 8*half, col +l.
    auto store_tile = [&](v8f c, int mbase, int nbase) {
#pragma unroll
        for (int r = 0; r < 8; ++r) {
            float v = 1.0f / (1.0f + __expf(-c[r]));
            scores[(size_t)(mbase + half * 8 + r) * N_EXPERTS + (nbase + l)] = v;
        }
    };
    store_tile(c00, m0,      n0);
    store_tile(c01, m0,      n0 + 16);
    store_tile(c10, m0 + 16, n0);
    store_tile(c11, m0 + 16, n0 + 16);
}

// ---------------------------------------------------------------------------
// Grouped top-k routing. block = 128 (4 waves), one wave32 per token.
// Lane l owns experts [8l, 8l+8); group g = l>>2 (4 lanes * 8 = 32 experts).
// Outputs: out[0 .. n*8)          = routing weights (fp32)
//          out[n*8 .. n*16)       = expert indices (as fp32 values)
// ---------------------------------------------------------------------------
__global__ __launch_bounds__(128)
void gate_topk(const float* __restrict__ scores,
               const float* __restrict__ bias,
               float* __restrict__ out,
               int nTokens)
{
    const int lane  = threadIdx.x & 31;
    const int wave  = threadIdx.x >> 5;
    const int token = blockIdx.x * 4 + wave;
    if (token >= nTokens) return;

    const float NEG = -__builtin_inff();
    const float* s  = scores + (size_t)token * N_EXPERTS;

    const int g     = lane >> 2;       // group 0..7
    const int ebase = lane << 3;       // first expert owned by this lane

    float sv[8];   // original sigmoid scores (gather source)
    float rv[8];   // ranking scores = sigmoid + bias
#pragma unroll
    for (int j = 0; j < 8; ++j) {
        sv[j] = s[ebase + j];
        rv[j] = sv[j] + bias[ebase + j];
    }

    // ---- per-lane top-2, then quad-merge -> group score = sum of top-2 ----
    float m1 = NEG, m2 = NEG;
#pragma unroll
    for (int j = 0; j < 8; ++j) {
        float t = rv[j];
        if (t > m1) { m2 = m1; m1 = t; }
        else if (t > m2) { m2 = t; }
    }
#pragma unroll
    for (int off = 1; off <= 2; off <<= 1) {
        float o1 = __shfl_xor(m1, off, 32);
        float o2 = __shfl_xor(m2, off, 32);
        float hi = fmaxf(m1, o1);
        float lo = fmaxf(fminf(m1, o1), fmaxf(m2, o2));
        m1 = hi; m2 = lo;
    }
    const float gscore = m1 + m2;

    // ---- broadcast all 8 group scores; redundant top-4 group selection ----
    float gs[8];
#pragma unroll
    for (int j = 0; j < 8; ++j)
        gs[j] = __shfl(gscore, j * 4, 32);

    unsigned keep = 0;
    for (int it = 0; it < TOPK_GROUPS; ++it) {
        float best = NEG; int bg = 0;
#pragma unroll
        for (int j = 0; j < 8; ++j)
            if (gs[j] >= best) { best = gs[j]; bg = j; }   // ties -> larger idx
        keep |= (1u << bg);
        gs[bg] = NEG;
    }
    if (!((keep >> g) & 1u)) {
#pragma unroll
        for (int j = 0; j < 8; ++j) rv[j] = NEG;
    }

    // ---- 8 rounds of wave-wide argmax (descending), tie -> larger index ----
    int   idxs[TOPK];
    float wts[TOPK];
    for (int it = 0; it < TOPK; ++it) {
        float bv = NEG; int bi = -1; float bs = 0.0f;
#pragma unroll
        for (int j = 0; j < 8; ++j)
            if (rv[j] >= bv) { bv = rv[j]; bi = ebase + j; bs = sv[j]; }
#pragma unroll
        for (int off = 16; off >= 1; off >>= 1) {
            float ov = __shfl_xor(bv, off, 32);
            int   oi = __shfl_xor(bi, off, 32);
            float os = __shfl_xor(bs, off, 32);
            if (ov > bv || (ov == bv && oi > bi)) { bv = ov; bi = oi; bs = os; }
        }
        idxs[it] = bi;
        wts[it]  = bs;
#pragma unroll
        for (int j = 0; j < 8; ++j)
            if (ebase + j == bi) rv[j] = NEG;   // owner clears winning slot
    }

    // ---- normalize and write (reference order = ascending score) ----
    float sum = 0.0f;
#pragma unroll
    for (int it = 0; it < TOPK; ++it) sum += wts[it];
    const float scale = ROUTE_SCALE / (sum + 1e-20f);

    if (lane < TOPK) {
        float wv = 0.0f; int iv = 0;
#pragma unroll
        for (int it = 0; it < TOPK; ++it)
            if (lane == (TOPK - 1) - it) { wv = wts[it]; iv = idxs[it]; }
        out[(size_t)token * TOPK + lane] = wv * scale;
        out[(size_t)nTokens * TOPK + (size_t)token * TOPK + lane] = (float)iv;
    }
}

// ---------------------------------------------------------------------------
extern "C" void kernel_launch(void* const* d_in, const int* in_sizes, int n_in,
                              void* d_out, int out_size, void* d_ws, size_t ws_size,
                              hipStream_t stream)
{
    const float* x    = (const float*)d_in[0];   // [n, 4096]
    const float* w    = (const float*)d_in[1];   // [256, 4096]
    const float* bias = (const float*)d_in[2];   // [256]
    float* out        = (float*)d_out;
    float* scores     = (float*)d_ws;            // [n, 256] fp32 (16 MB)

    const int nTokens = in_sizes[0] / DIM;

    dim3 gGemm(nTokens / 32);
    gate_gemm_sigmoid<<<gGemm, 256, 0, stream>>>(x, w, scores);

    dim3 gTopk((nTokens + 3) / 4);
    gate_topk<<<gTopk, 128, 0, stream>>>(scores, bias, out, nTokens);
}